// Edge2NodeProp_26912265077099
// MI455X (gfx1250) — compile-verified
//
#include <hip/hip_runtime.h>
#include <stdint.h>

typedef float v2f __attribute__((ext_vector_type(2)));
typedef float v8f __attribute__((ext_vector_type(8)));

#define EDGE_DIM 128
#define N_RADIAL 16
#define N_EDGES  640000
#define N_NODES  40000

// Full-precision matrix op: D(16x16,f32) = A(16x4,f32) x B(4x16,f32) + C
// A: lane<16 holds A[lane][k0..k0+1], lane>=16 holds A[lane-16][k0+2..k0+3]
// B: lane float2 = (B[k0][n], B[k0+1][n]) with n = lane%16, k0 += 2 for upper half
// D: VGPR r -> row r + 8*(lane/16), col = lane%16
__device__ __forceinline__ v8f wmma_16x16x4(v2f a, v2f b, v8f c) {
  return __builtin_amdgcn_wmma_f32_16x16x4_f32(false, a, false, b, (short)0, c,
                                               false, false);
}

__device__ __forceinline__ void atomic_add_f32_global(float* p, float v) {
  // Native L2 fp32 atomic (STOREcnt, no return). No "memory" clobber: the
  // destination (acc_nodes) is disjoint from everything this kernel loads,
  // so surrounding loads may be freely scheduled around it.
  asm volatile("global_atomic_add_f32 %0, %1, off" : : "v"(p), "v"(v));
}

__global__ __launch_bounds__(256) void zero_kernel(float4* __restrict__ p, int n4) {
  int i = blockIdx.x * blockDim.x + threadIdx.x;
  int stride = gridDim.x * blockDim.x;
  float4 z = {0.f, 0.f, 0.f, 0.f};
  for (; i < n4; i += stride) p[i] = z;
}

// -------------------- Edge pass: h = (rbf @ Wrbf^T) * x, scatter-add --------
// One wave handles a tile of 16 edges; 8 WMMA accumulators cover all 128 dims.
__global__ __launch_bounds__(128) void edge_kernel(
    const float* __restrict__ x, const float* __restrict__ rbf,
    const long long* __restrict__ idx, const float* __restrict__ Wrbf,
    float* __restrict__ acc_nodes) {
  const int lane = threadIdx.x & 31;
  const int lr = lane & 15;
  const int half = lane >> 4;
  const int wave = (blockIdx.x * blockDim.x + threadIdx.x) >> 5;
  const int nwaves = (gridDim.x * blockDim.x) >> 5;

  // Preload B = Wrbf^T tiles into registers (invariant across all edges).
  v2f B[8][4];
#pragma unroll
  for (int t = 0; t < 8; ++t)
#pragma unroll
    for (int kk = 0; kk < 4; ++kk)
      B[t][kk] = *(const v2f*)(Wrbf + (16 * t + lr) * N_RADIAL + 4 * kk + 2 * half);

  const v8f vz = {0.f, 0.f, 0.f, 0.f, 0.f, 0.f, 0.f, 0.f};
  const int ntiles = N_EDGES / 16;
  for (int tile = wave; tile < ntiles; tile += nwaves) {
    const int e0 = tile * 16;
    // A = rbf tile (16 edges x 16 radial), K split over 4 chained WMMAs
    v2f A[4];
#pragma unroll
    for (int kk = 0; kk < 4; ++kk)
      A[kk] = *(const v2f*)(rbf + (long long)(e0 + lr) * N_RADIAL + 4 * kk + 2 * half);

    v8f acc[8];
#pragma unroll
    for (int t = 0; t < 8; ++t) {
      acc[t] = vz;
#pragma unroll
      for (int kk = 0; kk < 4; ++kk) acc[t] = wmma_16x16x4(A[kk], B[t][kk], acc[t]);
    }

    // Scatter: my lane owns edge rows r+8*half (r=0..7), column 16t+lr.
    int nodeid[8];
#pragma unroll
    for (int r = 0; r < 8; ++r) nodeid[r] = (int)idx[e0 + r + 8 * half];

#pragma unroll
    for (int r = 0; r < 8; ++r) {
      const float* xrow = x + (long long)(e0 + r + 8 * half) * EDGE_DIM + lr;
      // Batch the 8 x loads into independent temporaries -> one load clause.
      float xv[8];
#pragma unroll
      for (int t = 0; t < 8; ++t) xv[t] = xrow[16 * t];
      float* nrow = acc_nodes + (long long)nodeid[r] * EDGE_DIM + lr;
#pragma unroll
      for (int t = 0; t < 8; ++t)
        atomic_add_f32_global(nrow + 16 * t, acc[t][r] * xv[t]);
    }
  }
}

// -------------------- Node pass: 3x (dense 128x128 + SiLU) + out proj -------
// One wave per 32 nodes (two 16-row A tiles share every B fetch -> halves the
// weight traffic). Features staged in LDS with padded stride (conflict-free).
#define FS 132  // padded LDS row stride (floats)

__global__ __launch_bounds__(64) void node_kernel(
    const float* __restrict__ feat_in, const float* __restrict__ W1,
    const float* __restrict__ b1, const float* __restrict__ W2,
    const float* __restrict__ b2, const float* __restrict__ W3,
    const float* __restrict__ b3, const float* __restrict__ Wout,
    float* __restrict__ out) {
  __shared__ float ldsF[2 * 32 * FS];  // 2 waves/WG, 32x128 feature tile each
  const int lane = threadIdx.x & 31;
  const int lr = lane & 15;
  const int half = lane >> 4;
  const int wv = threadIdx.x >> 5;
  float* F = ldsF + wv * 32 * FS;
  const int n0 = (blockIdx.x * 2 + wv) * 32;  // first node of this wave

  // Load 32 node rows (coalesced) into LDS.
  for (int r = 0; r < 32; ++r)
    for (int c = lane; c < EDGE_DIM; c += 32)
      F[r * FS + c] = feat_in[(long long)(n0 + r) * EDGE_DIM + c];
  // Same-wave LDS ops are in-order (DScnt); no cross-wave sharing -> no barrier.

  const float* Ws[3] = {W1, W2, W3};
  const float* bs[3] = {b1, b2, b3};
  const v8f vz = {0.f, 0.f, 0.f, 0.f, 0.f, 0.f, 0.f, 0.f};

  for (int layer = 0; layer < 3; ++layer) {
    const float* W = Ws[layer];
    const float* b = bs[layer];
    v8f acc0[8], acc1[8];
#pragma unroll
    for (int t = 0; t < 8; ++t) { acc0[t] = vz; acc1[t] = vz; }

    for (int kk = 0; kk < 32; ++kk) {  // K = 128 in steps of 4
      v2f a0 = *(const v2f*)(F + lr * FS + 4 * kk + 2 * half);
      v2f a1 = *(const v2f*)(F + (16 + lr) * FS + 4 * kk + 2 * half);
#pragma unroll
      for (int t = 0; t < 8; ++t) {
        // B = W^T slice: (W[16t+lr][k0], W[16t+lr][k0+1]); hits WGP$/L2
        v2f bb = *(const v2f*)(W + (16 * t + lr) * EDGE_DIM + 4 * kk + 2 * half);
        acc0[t] = wmma_16x16x4(a0, bb, acc0[t]);
        acc1[t] = wmma_16x16x4(a1, bb, acc1[t]);
      }
    }

    // bias + SiLU applied in D layout, written back row-major into LDS
#pragma unroll
    for (int t = 0; t < 8; ++t) {
      const int col = 16 * t + lr;
      const float bias = b[col];
#pragma unroll
      for (int r = 0; r < 8; ++r) {
        float v0 = acc0[t][r] + bias;
        v0 = v0 / (1.0f + __expf(-v0));  // silu
        F[(r + 8 * half) * FS + col] = v0;
        float v1 = acc1[t][r] + bias;
        v1 = v1 / (1.0f + __expf(-v1));
        F[(16 + r + 8 * half) * FS + col] = v1;
      }
    }
  }

  // Final projection to 1 dim: each lane owns one of the 32 nodes.
  float s = 0.f;
  for (int j = 0; j < EDGE_DIM; ++j) s += F[lane * FS + j] * Wout[j];
  out[n0 + lane] = s;
}

extern "C" void kernel_launch(void* const* d_in, const int* in_sizes, int n_in,
                              void* d_out, int out_size, void* d_ws, size_t ws_size,
                              hipStream_t stream) {
  (void)in_sizes; (void)n_in; (void)out_size; (void)ws_size;
  const float* x = (const float*)d_in[0];
  const float* rbf = (const float*)d_in[1];
  const long long* idx = (const long long*)d_in[2];  // int64 in reference
  // d_in[3] = num_nodes (compile-time constant here)
  const float* Wrbf = (const float*)d_in[4];
  const float* W1 = (const float*)d_in[5];
  const float* b1 = (const float*)d_in[6];
  const float* W2 = (const float*)d_in[7];
  const float* b2 = (const float*)d_in[8];
  const float* W3 = (const float*)d_in[9];
  const float* b3 = (const float*)d_in[10];
  const float* Wout = (const float*)d_in[11];
  float* out = (float*)d_out;
  float* acc = (float*)d_ws;  // N_NODES * 128 floats = 20.5 MB scratch

  const int n4 = N_NODES * EDGE_DIM / 4;
  zero_kernel<<<2048, 256, 0, stream>>>((float4*)acc, n4);
  edge_kernel<<<2048, 128, 0, stream>>>(x, rbf, idx, Wrbf, acc);
  node_kernel<<<N_NODES / 64, 64, 0, stream>>>(acc, W1, b1, W2, b2, W3, b3, Wout, out);
}